// BPR_54322746360498
// MI455X (gfx1250) — compile-verified
//
#include <hip/hip_runtime.h>

typedef __attribute__((ext_vector_type(2))) float v2f;
typedef __attribute__((ext_vector_type(8))) float v8f;

#define EMBED_DIM     64
#define MAX_FRIENDS   32
#define ROWS_PER_BLK  16
#define LDS_STRIDE    68   // 64 + 4 pad: keeps WMMA operand loads bank-conflict-free (64-bank LDS)

__global__ __launch_bounds__(512)
void bpr_social_kernel(const int*   __restrict__ users,
                       const int*   __restrict__ items,
                       const int*   __restrict__ friends,
                       const int*   __restrict__ friend_counts,
                       const float* __restrict__ user_emb,
                       const float* __restrict__ item_emb,
                       const float* __restrict__ user_friend,
                       float*       __restrict__ out,
                       int batch, int num_users)
{
    __shared__ float Vlds [ROWS_PER_BLK * LDS_STRIDE];   // combined vectors v[b] = ue + sum_f w_f*fe_f
    __shared__ float IElds[ROWS_PER_BLK * LDS_STRIDE];   // item embeddings

    const int lane = threadIdx.x & 31;
    const int wave = threadIdx.x >> 5;                   // 16 waves, one batch row each
    const int b    = blockIdx.x * ROWS_PER_BLK + wave;

    // ---------------- gather + weighted-friend-accumulation phase ----------------
    if (b < batch) {
        const int u   = users[b];
        const int it  = items[b];
        const int cnt = friend_counts[u];
        const float denom = (float)((cnt < 1) ? 1 : cnt);

        // lane f owns friend f: coalesced friend-index load + the single scatter into user_friend
        const int frf = friends[u * MAX_FRIENDS + lane];
        float wt = 0.0f;
        if (lane < cnt)
            wt = user_friend[(size_t)u * (size_t)num_users + (size_t)frf] / denom;

        // lane holds dims {lane, lane+32}
        float v_lo = user_emb[(size_t)u * EMBED_DIM + lane];
        float v_hi = user_emb[(size_t)u * EMBED_DIM + lane + 32];
        const float ie_lo = item_emb[(size_t)it * EMBED_DIM + lane];
        const float ie_hi = item_emb[(size_t)it * EMBED_DIM + lane + 32];

        // accumulate v += w_f * fe_f ; broadcast (w,idx) per friend so fe loads are coalesced 256B rows
        #pragma unroll 8
        for (int f = 0; f < MAX_FRIENDS; ++f) {
            const float wf = __shfl(wt, f, 32);
            if (wf != 0.0f) {                       // wave-uniform: skips masked friends entirely
                const int ff = __shfl(frf, f, 32);
                v_lo += wf * user_emb[(size_t)ff * EMBED_DIM + lane];
                v_hi += wf * user_emb[(size_t)ff * EMBED_DIM + lane + 32];
            }
        }

        Vlds [wave * LDS_STRIDE + lane]      = v_lo;
        Vlds [wave * LDS_STRIDE + lane + 32] = v_hi;
        IElds[wave * LDS_STRIDE + lane]      = ie_lo;
        IElds[wave * LDS_STRIDE + lane + 32] = ie_hi;
    } else {
        // keep WMMA inputs defined for tail blocks
        Vlds [wave * LDS_STRIDE + lane]      = 0.0f;
        Vlds [wave * LDS_STRIDE + lane + 32] = 0.0f;
        IElds[wave * LDS_STRIDE + lane]      = 0.0f;
        IElds[wave * LDS_STRIDE + lane + 32] = 0.0f;
    }
    __syncthreads();

    // ---------------- WMMA phase: out[m] = diag( V(16x64) x IE^T(64x16) ) ----------------
    if (wave == 0) {
        // f32 A(16x4) layout: lane L -> M=L%16 ; VGPR0 = K=(L<16?0:2), VGPR1 = K=(L<16?1:3)
        // f32 B(4x16) layout: lane L -> N=L%16 ; VGPR0 = K=(L<16?0:2), VGPR1 = K=(L<16?1:3)
        const int row  = lane & 15;
        const int koff = (lane >> 4) << 1;   // 0 for lanes 0-15, 2 for lanes 16-31
        const float* vp = &Vlds [row * LDS_STRIDE + koff];
        const float* bp = &IElds[row * LDS_STRIDE + koff];

        v8f acc = {};
        #pragma unroll
        for (int k0 = 0; k0 < EMBED_DIM; k0 += 4) {
            v2f a;  a.x  = vp[k0]; a.y  = vp[k0 + 1];
            v2f bb; bb.x = bp[k0]; bb.y = bp[k0 + 1];
            acc = __builtin_amdgcn_wmma_f32_16x16x4_f32(
                /*neg_a=*/false, a, /*neg_b=*/false, bb,
                /*c_mod=*/(short)0, acc, /*reuse_a=*/false, /*reuse_b=*/false);
        }

        // diagonal extraction from 16x16 f32 C/D layout:
        //   m = 0..7  : VGPR m    at lane m
        //   m = 8..15 : VGPR m-8  at lane m+16
        int m = 0, idx = 0;
        bool active = false;
        if (lane < 8)        { m = lane;      idx = lane;      active = true; }
        else if (lane >= 24) { m = lane - 16; idx = lane - 24; active = true; }
        if (active) {
            const float val =
                (idx == 0) ? acc[0] :
                (idx == 1) ? acc[1] :
                (idx == 2) ? acc[2] :
                (idx == 3) ? acc[3] :
                (idx == 4) ? acc[4] :
                (idx == 5) ? acc[5] :
                (idx == 6) ? acc[6] : acc[7];
            const int ob = blockIdx.x * ROWS_PER_BLK + m;
            if (ob < batch) out[ob] = val;
        }
    }
}

extern "C" void kernel_launch(void* const* d_in, const int* in_sizes, int n_in,
                              void* d_out, int out_size, void* d_ws, size_t ws_size,
                              hipStream_t stream) {
    const int*   users         = (const int*)  d_in[0];
    const int*   items         = (const int*)  d_in[1];
    const int*   friends       = (const int*)  d_in[2];
    const int*   friend_counts = (const int*)  d_in[3];
    const float* user_emb      = (const float*)d_in[4];
    const float* item_emb      = (const float*)d_in[5];
    const float* user_friend   = (const float*)d_in[6];
    float*       out           = (float*)d_out;

    const int batch     = in_sizes[0];
    const int num_users = in_sizes[3];           // friend_counts has NUM_USERS entries
    const int blocks    = (batch + ROWS_PER_BLK - 1) / ROWS_PER_BLK;

    bpr_social_kernel<<<blocks, 512, 0, stream>>>(
        users, items, friends, friend_counts,
        user_emb, item_emb, user_friend, out, batch, num_users);
}